// VariationalGCN_11854109737492
// MI455X (gfx1250) — compile-verified
//
#include <hip/hip_runtime.h>

#define N_NODES 50000
#define N_EDGES 800000

typedef __attribute__((ext_vector_type(2))) float v2f;
typedef __attribute__((ext_vector_type(8))) float v8f;

// ---------------- utility kernels ----------------

__global__ void fill_zero(float* __restrict__ p, long n) {
  long i = (long)blockIdx.x * blockDim.x + threadIdx.x;
  if (i < n) p[i] = 0.0f;
}

__global__ void deg_kernel(const long long* __restrict__ dst,
                           const float* __restrict__ ew,
                           float* __restrict__ deg, int E) {
  int e = blockIdx.x * blockDim.x + threadIdx.x;
  if (e < E) atomicAdd(&deg[(size_t)dst[e]], ew[e]);
}

__global__ void dinv_kernel(float* __restrict__ deg, int n) {
  int i = blockIdx.x * blockDim.x + threadIdx.x;
  if (i < n) deg[i] = rsqrtf(deg[i] + 1.0f);   // deg includes +1 self-loop, always > 0
}

__global__ void norm_kernel(const long long* __restrict__ src,
                            const long long* __restrict__ dst,
                            const float* __restrict__ ew,
                            const float* __restrict__ dinv,
                            float* __restrict__ norm, int E) {
  int e = blockIdx.x * blockDim.x + threadIdx.x;
  if (e < E) norm[e] = dinv[(size_t)src[e]] * ew[e] * dinv[(size_t)dst[e]];
}

// ---------------- dense GEMM: H = A(Nx128) @ W(128xDOUT), fp32 WMMA ----------------
// A-stationary: one wave32 computes the full 16 x DOUT output panel.
//  - A 16x4 fragment loaded ONCE per K-step (1x global_load_b64/lane), reused
//    across all DOUT/16 accumulators -> 8x less A traffic vs tile-per-wave.
//  - NT independent accumulator chains give the fp32 WMMA pipe ILP to hide
//    latency inside a single wave.
//
// Layouts (wave32):
//  16x4  f32 A: lane L -> row (L&15), K-pair starting at 2*(L>>4)   (v2f)
//  4x16  f32 B: lane L -> col (L&15), same K-pair                   (v2f)
//  16x16 f32 D: vgpr v, lane L -> row v + 8*(L>>4), col (L&15)      (v8f)
template <int DOUT>
__global__ __launch_bounds__(32)
void gemm_wmma_f32(const float* __restrict__ A, const float* __restrict__ W,
                   float* __restrict__ H) {
  constexpr int NT = DOUT / 16;
  const int mt   = blockIdx.x;          // 0..3124 (50000/16 exactly)
  const int lane = threadIdx.x;         // 0..31, EXEC all-1s (WMMA requirement)
  const int r    = lane & 15;
  const int kh   = lane >> 4;           // K-pair half: 0 -> K0,K1 ; 1 -> K2,K3

  const float* Arow = A + (size_t)(mt * 16 + r) * 128 + kh * 2;
  const float* Wlan = W + (size_t)r;    // + nt*16 + k*DOUT below

  v8f acc[NT];
#pragma unroll
  for (int nt = 0; nt < NT; ++nt) acc[nt] = (v8f){};

#pragma unroll
  for (int kb = 0; kb < 128; kb += 4) {
    const float2 av = *(const float2*)(Arow + kb);
    v2f a;
    a.x = av.x;
    a.y = av.y;
#pragma unroll
    for (int nt = 0; nt < NT; ++nt) {
      v2f b;
      b.x = Wlan[(size_t)(kb + kh * 2    ) * DOUT + nt * 16];
      b.y = Wlan[(size_t)(kb + kh * 2 + 1) * DOUT + nt * 16];
      acc[nt] = __builtin_amdgcn_wmma_f32_16x16x4_f32(
          /*neg_a=*/false, a, /*neg_b=*/false, b,
          /*c_mod=*/(short)0, acc[nt], /*reuse_a=*/false, /*reuse_b=*/false);
    }
  }

  float* Hout = H + (size_t)(mt * 16 + 8 * kh) * DOUT + r;
#pragma unroll
  for (int nt = 0; nt < NT; ++nt)
#pragma unroll
    for (int v = 0; v < 8; ++v)
      Hout[(size_t)v * DOUT + nt * 16] = acc[nt][v];
}

// ---------------- edge scatter: agg[dst] += h[src] * norm ----------------
// One thread per (edge, feature). D is a power of two so a wave32 covers 32
// consecutive features of one edge: coalesced gather + coalesced f32 atomics
// (edge-scalar src/dst/norm loads are same-address within the wave and
// collapse to a single request).
template <int D>
__global__ void scatter_kernel(const float* __restrict__ h,
                               const float* __restrict__ norm,
                               const long long* __restrict__ src,
                               const long long* __restrict__ dst,
                               float* __restrict__ agg) {
  long idx = (long)blockIdx.x * blockDim.x + threadIdx.x;
  const long total = (long)N_EDGES * D;
  if (idx >= total) return;
  const int e = (int)(idx >> __builtin_ctz(D));
  const int f = (int)(idx & (D - 1));
  const float v = h[(size_t)src[e] * D + f] * norm[e];
  atomicAdd(&agg[(size_t)dst[e] * D + f], v);
}

// ---------------- combine: out = agg + h*dinv^2 + b  (optional ReLU) ----------------
template <int D, bool RELU>
__global__ void combine_kernel(const float* __restrict__ agg,
                               const float* __restrict__ h,
                               const float* __restrict__ dinv,
                               const float* __restrict__ b,
                               float* __restrict__ out) {
  long idx = (long)blockIdx.x * blockDim.x + threadIdx.x;
  const long total = (long)N_NODES * D;
  if (idx >= total) return;
  const int i = (int)(idx >> __builtin_ctz(D));
  const int f = (int)(idx & (D - 1));
  const float d = dinv[i];
  float v = agg[idx] + h[idx] * d * d + b[f];
  if (RELU) v = fmaxf(v, 0.0f);
  out[idx] = v;
}

// ---------------- driver ----------------

extern "C" void kernel_launch(void* const* d_in, const int* in_sizes, int n_in,
                              void* d_out, int out_size, void* d_ws, size_t ws_size,
                              hipStream_t stream) {
  const float*     x   = (const float*)d_in[0];
  const long long* ei  = (const long long*)d_in[1];   // int64 [2, E]
  const float*     ew  = (const float*)d_in[2];
  const float*     W1  = (const float*)d_in[3];
  const float*     b1  = (const float*)d_in[4];
  const float*     W2  = (const float*)d_in[5];
  const float*     b2  = (const float*)d_in[6];
  const float*     Wmu = (const float*)d_in[7];
  const float*     bmu = (const float*)d_in[8];
  const float*     Wst = (const float*)d_in[9];
  const float*     bst = (const float*)d_in[10];

  const long long* src = ei;
  const long long* dst = ei + N_EDGES;
  float* out = (float*)d_out;                         // [mu(N*64) | std(N*64)]

  // workspace layout (floats)
  float* ws   = (float*)d_ws;
  float* dinv = ws;                                   // N (deg -> dinv in place)
  float* norm = dinv + N_NODES;                       // E
  float* h    = norm + N_EDGES;                       // N*128 (gemm output)
  float* agg  = h + (size_t)N_NODES * 128;            // N*128 (scatter accum)
  float* hb   = agg + (size_t)N_NODES * 128;          // N*128 (layer activations)

  const long NH  = (long)N_NODES * 128;
  const long NO  = (long)N_NODES * 64;
  const long EH  = (long)N_EDGES * 128;
  const long EO  = (long)N_EDGES * 64;
  const int  TB  = 256;
  auto blk = [](long n, int tb) { return (unsigned)((n + tb - 1) / tb); };

  // ---- degree / normalization (shared by all 4 convs) ----
  fill_zero<<<blk(N_NODES, TB), TB, 0, stream>>>(dinv, N_NODES);
  deg_kernel<<<blk(N_EDGES, TB), TB, 0, stream>>>(dst, ew, dinv, N_EDGES);
  dinv_kernel<<<blk(N_NODES, TB), TB, 0, stream>>>(dinv, N_NODES);
  norm_kernel<<<blk(N_EDGES, TB), TB, 0, stream>>>(src, dst, ew, dinv, norm, N_EDGES);

  const unsigned gM = N_NODES / 16;   // 3125 panel-waves

  // ---- layer 1: hb = relu(gcn(x, W1, b1)) ----
  gemm_wmma_f32<128><<<gM, 32, 0, stream>>>(x, W1, h);
  fill_zero<<<blk(NH, TB), TB, 0, stream>>>(agg, NH);
  scatter_kernel<128><<<blk(EH, TB), TB, 0, stream>>>(h, norm, src, dst, agg);
  combine_kernel<128, true><<<blk(NH, TB), TB, 0, stream>>>(agg, h, dinv, b1, hb);

  // ---- layer 2: hb = relu(gcn(hb, W2, b2)) ----
  gemm_wmma_f32<128><<<gM, 32, 0, stream>>>(hb, W2, h);
  fill_zero<<<blk(NH, TB), TB, 0, stream>>>(agg, NH);
  scatter_kernel<128><<<blk(EH, TB), TB, 0, stream>>>(h, norm, src, dst, agg);
  combine_kernel<128, true><<<blk(NH, TB), TB, 0, stream>>>(agg, h, dinv, b2, hb);

  // ---- mu = gcn(hb, Wmu, bmu) -> out[0 : N*64) ----
  gemm_wmma_f32<64><<<gM, 32, 0, stream>>>(hb, Wmu, h);
  fill_zero<<<blk(NO, TB), TB, 0, stream>>>(agg, NO);
  scatter_kernel<64><<<blk(EO, TB), TB, 0, stream>>>(h, norm, src, dst, agg);
  combine_kernel<64, false><<<blk(NO, TB), TB, 0, stream>>>(agg, h, dinv, bmu, out);

  // ---- std = gcn(hb, Wstd, bstd) -> out[N*64 : 2*N*64) ----
  gemm_wmma_f32<64><<<gM, 32, 0, stream>>>(hb, Wst, h);
  fill_zero<<<blk(NO, TB), TB, 0, stream>>>(agg, NO);
  scatter_kernel<64><<<blk(EO, TB), TB, 0, stream>>>(h, norm, src, dst, agg);
  combine_kernel<64, false><<<blk(NO, TB), TB, 0, stream>>>(agg, h, dinv, bst, out + NO);
}